// SimilarityTripletLoss_16655883174498
// MI455X (gfx1250) — compile-verified
//
#include <hip/hip_runtime.h>
#include <math.h>

typedef __attribute__((ext_vector_type(2))) float v2f;
typedef __attribute__((ext_vector_type(8))) float v8f;
typedef __attribute__((ext_vector_type(4))) int   v4i;

#define BDIM   16
#define CDIM   1024
#define HW     1024
#define FWI    32
#define TM     128
#define TN     64
#define TK     16
#define NKT    (CDIM / TK)        // 64 K-steps
#define NTN    (HW / TN)          // 16 column tiles
#define NTM    (HW / TM)          // 8 row tiles
#define NPAIR  (BDIM * HW)        // 16384 (b, n) pairs
#define MARGINF 0.6f
#define EPSF    1e-8f
#define BIGF    3.4e38f

// workspace layout (float elements)
#define WS_ANORM 0
#define WS_RNORM (WS_ANORM + NPAIR)
#define WS_CAP   (WS_RNORM + NPAIR)                 // cos_ap per (b,n)
#define WS_PART  (WS_CAP + NPAIR)                   // 3 minima per (b,n,tile_n)
#define WS_BSUM  (WS_PART + NPAIR * NTN * 3)        // 64 block sums

// --------------------------- async global -> LDS ---------------------------
// Builtin params (from round-3 diagnostics): b32 takes (__device__ int*,
// __shared__ int*, imm, imm); b128 takes the int4-typed equivalents.
__device__ __forceinline__ void async_g2l_b32(const float* g, float* l) {
#if __has_builtin(__builtin_amdgcn_global_load_async_to_lds_b32)
    __builtin_amdgcn_global_load_async_to_lds_b32(
        (__attribute__((address_space(1))) int*)(g),
        (__attribute__((address_space(3))) int*)(l), 0, 0);
#else
    unsigned lo = (unsigned)(size_t)l;     // low 32 bits of generic = LDS offset
    asm volatile("global_load_async_to_lds_b32 %0, %1, off"
                 :: "v"(lo), "v"(g) : "memory");
#endif
}

__device__ __forceinline__ void async_g2l_b128(const float* g, float* l) {
#if __has_builtin(__builtin_amdgcn_global_load_async_to_lds_b128)
    __builtin_amdgcn_global_load_async_to_lds_b128(
        (__attribute__((address_space(1))) v4i*)(g),
        (__attribute__((address_space(3))) v4i*)(l), 0, 0);
#else
    unsigned lo = (unsigned)(size_t)l;
    asm volatile("global_load_async_to_lds_b128 %0, %1, off"
                 :: "v"(lo), "v"(g) : "memory");
#endif
}

#if __has_builtin(__builtin_amdgcn_s_wait_asynccnt)
#define WAIT_ASYNC(n) __builtin_amdgcn_s_wait_asynccnt(n)
#else
#define WAIT_ASYNC(n) asm volatile("s_wait_asynccnt %0" :: "n"(n) : "memory")
#endif

// ---------------------------------------------------------------------------
// L2 norm over C of every spatial column: x is (B, C, HW); one thread per (b,n)
// ---------------------------------------------------------------------------
__global__ __launch_bounds__(256) void norms_kernel(const float* __restrict__ x,
                                                    float* __restrict__ out) {
    int idx = blockIdx.x * 256 + threadIdx.x;       // 0..16383
    int b = idx >> 10, n = idx & 1023;
    const float* p = x + (size_t)b * CDIM * HW + n;
    float s = 0.0f;
#pragma unroll 4
    for (int c = 0; c < CDIM; ++c) {
        float v = p[(size_t)c * HW];
        s = fmaf(v, v, s);
    }
    out[idx] = sqrtf(s);
}

// ---------------------------------------------------------------------------
// Per-batch cosine-similarity tile (TM x TN) via V_WMMA_F32_16X16X4_F32.
// Block: 256 threads = 8 waves as 2 (M) x 4 (N); wave owns 64 rows x 16 cols.
// Double-buffered LDS staging fed by GLOBAL_LOAD_ASYNC_TO_LDS (ASYNCcnt),
// next tile issued before the WMMA burst of the current one.
// ---------------------------------------------------------------------------
__global__ __launch_bounds__(256) void simtile_kernel(
    const float* __restrict__ A,        // sketch (B, C, HW)
    const float* __restrict__ R,        // ref    (B, C, HW)
    const int*   __restrict__ aidx,     // (N, 3) = (b, x, y)
    const int*   __restrict__ pidx,     // (N, 3) = (b, x, y)
    const float* __restrict__ anorm,    // spatial norms of sketch
    const float* __restrict__ rnorm,    // spatial norms of ref
    float* __restrict__ partial,        // (B*HW, NTN, 3)
    float* __restrict__ cap)            // (B*HW) cos_ap
{
    __shared__ union U {
        struct {
            float a[TM][TK + 1];        // transposed A tile, padded
            float bt[TK][TN];           // B tile, K-major
        } st[2];                        // double buffer for async staging
        float sim[TM][TN + 1];          // normalized sim tile, padded rows
    } lds __attribute__((aligned(16)));
    __shared__ int aoff[TM];            // anchor spatial offset per local row
    __shared__ int poff[TM];            // positive spatial offset per local row

    const int tid  = threadIdx.x;
    const int lane = tid & 31;
    const int wid  = tid >> 5;
    const int wm   = wid >> 2;          // 0..1 : which 64-row slab
    const int wn   = wid & 3;           // 0..3 : which 16-col slab
    const int laneHalf = lane >> 4;     // 0|1 : K-pair select per ISA operand layout
    const int l15  = lane & 15;

    const int tn = blockIdx.x;          // 0..15
    const int tm = blockIdx.y;          // 0..7
    const int b  = blockIdx.z;          // 0..15

    const size_t bbase = (size_t)b * CDIM * HW;
    const float* Ab = A + bbase;
    const float* Rb = R + bbase + tn * TN;

    if (tid < TM) {
        int g  = b * HW + tm * TM + tid;
        aoff[tid] = aidx[g * 3 + 2] * FWI + aidx[g * 3 + 1];   // y*FW + x
        poff[tid] = pidx[g * 3 + 2] * FWI + pidx[g * 3 + 1];
    }
    __syncthreads();

    // Per-thread staging coordinates (K-invariant).
    const int sRow   = tid & 127;       // A-tile row this thread stages (b32 x8, gathered)
    const int sCC    = tid >> 7;        // 0|1 : base k-column (+2*i)
    const int sCCB4  = tid >> 4;        // 0..15 : B-tile k-row (one b128 each)
    const int sColB4 = (tid & 15) * 4;  // B-tile column group (16B aligned)
    const float* aSrc = Ab + aoff[sRow];
    const float* bSrc = Rb + sColB4;

    auto issueTile = [&](int kt, int buf) {
#pragma unroll
        for (int i = 0; i < 8; ++i)
            async_g2l_b32(&aSrc[(size_t)(kt * TK + sCC + 2 * i) * HW],
                          &lds.st[buf].a[sRow][sCC + 2 * i]);
        async_g2l_b128(&bSrc[(size_t)(kt * TK + sCCB4) * HW],
                       &lds.st[buf].bt[sCCB4][sColB4]);
    };

    v8f acc0 = {0, 0, 0, 0, 0, 0, 0, 0};
    v8f acc[4];
#pragma unroll
    for (int s = 0; s < 4; ++s) acc[s] = acc0;

    issueTile(0, 0);                    // 9 async ops in flight
    for (int kt = 0; kt < NKT; ++kt) {
        const int cur = kt & 1;
        if (kt + 1 < NKT) {
            issueTile(kt + 1, cur ^ 1); // 18 in flight
            WAIT_ASYNC(9);              // oldest 9 (= tile kt) complete, in order
        } else {
            WAIT_ASYNC(0);
        }
        __syncthreads();                // tile kt visible to all waves

        // Gather ALL fragments for this K=16 chunk, then fire 16 WMMAs.
        // ISA 32-bit operand layout: lanes 0-15 hold K = c0,c0+1 in VGPR0/1,
        // lanes 16-31 hold K = c0+2,c0+3  ->  folded via laneHalf*2.
        v2f afr[4][4], bfr[4];
#pragma unroll
        for (int kk = 0; kk < 4; ++kk) {
            const int c0 = kk * 4 + laneHalf * 2;
            bfr[kk].x = lds.st[cur].bt[c0][wn * 16 + l15];
            bfr[kk].y = lds.st[cur].bt[c0 + 1][wn * 16 + l15];
#pragma unroll
            for (int s = 0; s < 4; ++s) {
                const int row = wm * 64 + s * 16 + l15;
                afr[kk][s].x = lds.st[cur].a[row][c0];
                afr[kk][s].y = lds.st[cur].a[row][c0 + 1];
            }
        }
#pragma unroll
        for (int kk = 0; kk < 4; ++kk) {
#pragma unroll
            for (int s = 0; s < 4; ++s) {
                acc[s] = __builtin_amdgcn_wmma_f32_16x16x4_f32(
                    false, afr[kk][s], false, bfr[kk], (short)0, acc[s], false, false);
            }
        }
        __syncthreads();                // all reads done before buffer reuse
    }

    // Normalize and scatter accumulators into the LDS sim tile.
    // C/D layout: VGPR v -> row (v + laneHalf*8) of the 16x16 subtile, col l15.
    {
        const float* rn = rnorm + b * HW + tn * TN;
        const int colL = wn * 16 + l15;
        const float rv = rn[colL];
#pragma unroll
        for (int s = 0; s < 4; ++s) {
#pragma unroll
            for (int v = 0; v < 8; ++v) {
                const int rowL = wm * 64 + s * 16 + v + laneHalf * 8;
                const float an = anorm[b * HW + aoff[rowL]];
                lds.sim[rowL][colL] = acc[s][v] / fmaxf(an * rv, EPSF);
            }
        }
    }
    __syncthreads();

    // Per-row: 3 smallest sims in this column tile; cos_ap if positive col here.
    if (tid < TM) {
        float m1 = BIGF, m2 = BIGF, m3 = BIGF;
#pragma unroll 4
        for (int j = 0; j < TN; ++j) {
            float v = lds.sim[tid][j];
            if (v < m1)      { m3 = m2; m2 = m1; m1 = v; }
            else if (v < m2) { m3 = m2; m2 = v; }
            else if (v < m3) { m3 = v; }
        }
        const int rowG = tm * TM + tid;
        float* p = partial + ((size_t)(b * HW + rowG) * NTN + tn) * 3;
        p[0] = m1; p[1] = m2; p[2] = m3;
        const int pp = poff[tid];
        if ((pp >> 6) == tn)
            cap[b * HW + rowG] = lds.sim[tid][pp & 63];
    }
}

// ---------------------------------------------------------------------------
// Stage 1: merge 16 column-tile partials per (b,n), compute ReLU triplet loss,
// tree-reduce 256 losses per block -> 64 block sums. Deterministic.
// ---------------------------------------------------------------------------
__global__ __launch_bounds__(256) void loss_stage1(const float* __restrict__ partial,
                                                   const float* __restrict__ cap,
                                                   float* __restrict__ bsum) {
    __shared__ float red[256];
    int idx = blockIdx.x * 256 + threadIdx.x;       // 0..16383
    const float* p = partial + (size_t)idx * NTN * 3;
    float m1 = BIGF, m2 = BIGF, m3 = BIGF;
#pragma unroll 4
    for (int j = 0; j < NTN * 3; ++j) {
        float v = p[j];
        if (v < m1)      { m3 = m2; m2 = m1; m1 = v; }
        else if (v < m2) { m3 = m2; m2 = v; }
        else if (v < m3) { m3 = v; }
    }
    float loss = fmaxf(0.0f, (m1 + m2 + m3) - cap[idx] + MARGINF);
    red[threadIdx.x] = loss;
    __syncthreads();
    for (int s = 128; s > 0; s >>= 1) {
        if (threadIdx.x < s) red[threadIdx.x] += red[threadIdx.x + s];
        __syncthreads();
    }
    if (threadIdx.x == 0) bsum[blockIdx.x] = red[0];
}

// Stage 2: 64 block sums -> scalar mean.
__global__ __launch_bounds__(64) void loss_stage2(const float* __restrict__ bsum,
                                                  float* __restrict__ out) {
    __shared__ float red[64];
    red[threadIdx.x] = bsum[threadIdx.x];
    __syncthreads();
    for (int s = 32; s > 0; s >>= 1) {
        if (threadIdx.x < s) red[threadIdx.x] += red[threadIdx.x + s];
        __syncthreads();
    }
    if (threadIdx.x == 0) out[0] = red[0] / (1e-6f + (float)NPAIR);
}

// ---------------------------------------------------------------------------
extern "C" void kernel_launch(void* const* d_in, const int* in_sizes, int n_in,
                              void* d_out, int out_size, void* d_ws, size_t ws_size,
                              hipStream_t stream) {
    const float* sketch = (const float*)d_in[0];
    const float* ref    = (const float*)d_in[1];
    const int*   aidx   = (const int*)d_in[2];
    const int*   pidx   = (const int*)d_in[3];

    float* ws    = (float*)d_ws;
    float* anorm = ws + WS_ANORM;
    float* rnorm = ws + WS_RNORM;
    float* cap   = ws + WS_CAP;
    float* part  = ws + WS_PART;
    float* bsum  = ws + WS_BSUM;

    norms_kernel<<<NPAIR / 256, 256, 0, stream>>>(sketch, anorm);
    norms_kernel<<<NPAIR / 256, 256, 0, stream>>>(ref, rnorm);

    dim3 grid(NTN, NTM, BDIM);   // 16 x 8 x 16 workgroups
    simtile_kernel<<<grid, 256, 0, stream>>>(sketch, ref, aidx, pidx,
                                             anorm, rnorm, part, cap);

    loss_stage1<<<NPAIR / 256, 256, 0, stream>>>(part, cap, bsum);
    loss_stage2<<<1, 64, 0, stream>>>(bsum, (float*)d_out);
}